// RANSACBlock_62062277427650
// MI455X (gfx1250) — compile-verified
//
#include <hip/hip_runtime.h>
#include <hip/hip_bf16.h>
#include <stdint.h>

#define BB        128
#define NP        8192
#define NRANSAC   15
#define NITERS    50
#define ERRTOL2   16.0f     // (4.0)^2
#define FCAL      388.0f
#define CUX       253.5f
#define CVY       196.5f

typedef __attribute__((ext_vector_type(2))) float v2f;
typedef __attribute__((ext_vector_type(8))) float v8f;

// ---------------------------------------------------------------- utilities

__device__ __forceinline__ unsigned hash32(unsigned x) {
    x ^= x >> 16; x *= 0x7feb352dU;
    x ^= x >> 15; x *= 0x846ca68bU;
    x ^= x >> 16;
    return x;
}

__device__ __forceinline__ float det3(const float* M) {
    return M[0]*(M[4]*M[8]-M[5]*M[7])
         - M[1]*(M[3]*M[8]-M[5]*M[6])
         + M[2]*(M[3]*M[7]-M[4]*M[6]);
}

// C(3x4) = A(3x4, implicit [0001]) * B(3x4, implicit [0001])
__device__ __forceinline__ void mul34(const float* A, const float* Bm, float* C) {
    #pragma unroll
    for (int i = 0; i < 3; i++) {
        #pragma unroll
        for (int j = 0; j < 4; j++) {
            float s = (j == 3) ? A[i*4+3] : 0.0f;
            #pragma unroll
            for (int k = 0; k < 3; k++) s += A[i*4+k] * Bm[k*4+j];
            C[i*4+j] = s;
        }
    }
}

// Rigid-alignment rotation from 3x3 cross-covariance W:
// SVD W = U S V^T via Jacobi eigen of W^T W, R = U diag(1,1,det(U)det(V)) V^T
__device__ void svd_rigid(const float* W, float* R) {
    float A[9];
    #pragma unroll
    for (int i = 0; i < 3; i++)
        #pragma unroll
        for (int j = 0; j < 3; j++) {
            float s = 0.f;
            #pragma unroll
            for (int k = 0; k < 3; k++) s += W[k*3+i] * W[k*3+j];
            A[i*3+j] = s;
        }
    float V[9] = {1,0,0, 0,1,0, 0,0,1};
    for (int sweep = 0; sweep < 10; sweep++) {
        #pragma unroll
        for (int p = 0; p < 2; p++)
            #pragma unroll
            for (int q2 = 0; q2 < 3; q2++) {
                int q = q2;
                if (q <= p) continue;
                float apq = A[p*3+q];
                if (fabsf(apq) < 1e-25f) continue;
                float app = A[p*3+p], aqq = A[q*3+q];
                float tau = (aqq - app) / (2.0f * apq);
                float tt  = (tau >= 0.f ? 1.0f : -1.0f) /
                            (fabsf(tau) + sqrtf(1.0f + tau*tau));
                float c = rsqrtf(1.0f + tt*tt);
                float s = tt * c;
                #pragma unroll
                for (int k = 0; k < 3; k++) {
                    float akp = A[k*3+p], akq = A[k*3+q];
                    A[k*3+p] = c*akp - s*akq;
                    A[k*3+q] = s*akp + c*akq;
                }
                #pragma unroll
                for (int k = 0; k < 3; k++) {
                    float apk = A[p*3+k], aqk = A[q*3+k];
                    A[p*3+k] = c*apk - s*aqk;
                    A[q*3+k] = s*apk + c*aqk;
                }
                #pragma unroll
                for (int k = 0; k < 3; k++) {
                    float vkp = V[k*3+p], vkq = V[k*3+q];
                    V[k*3+p] = c*vkp - s*vkq;
                    V[k*3+q] = s*vkp + c*vkq;
                }
            }
    }
    float e[3] = {A[0], A[4], A[8]};
    // sort eigenpairs descending (columns of V)
    #pragma unroll
    for (int a = 0; a < 2; a++)
        #pragma unroll
        for (int bb = 0; bb < 2; bb++) {
            int j = bb;
            if (e[j+1] > e[j]) {
                float tmp = e[j]; e[j] = e[j+1]; e[j+1] = tmp;
                #pragma unroll
                for (int k = 0; k < 3; k++) {
                    float tv = V[k*3+j]; V[k*3+j] = V[k*3+j+1]; V[k*3+j+1] = tv;
                }
            }
        }
    float U[9];
    #pragma unroll
    for (int j = 0; j < 3; j++) {
        float sv  = sqrtf(fmaxf(e[j], 0.0f));
        float inv = (sv > 1e-12f) ? 1.0f / sv : 0.0f;
        #pragma unroll
        for (int i = 0; i < 3; i++) {
            float u = 0.f;
            #pragma unroll
            for (int k = 0; k < 3; k++) u += W[i*3+k] * V[k*3+j];
            U[i*3+j] = u * inv;
        }
    }
    // rebuild degenerate third left-singular vector
    float n2 = U[2]*U[2] + U[5]*U[5] + U[8]*U[8];
    if (n2 < 1e-8f) {
        float ax=U[0], ay=U[3], az=U[6], bx=U[1], by=U[4], bz=U[7];
        float cx=ay*bz-az*by, cy=az*bx-ax*bz, cz=ax*by-ay*bx;
        float inv = rsqrtf(fmaxf(cx*cx+cy*cy+cz*cz, 1e-30f));
        U[2]=cx*inv; U[5]=cy*inv; U[8]=cz*inv;
    }
    float d = (det3(U) * det3(V) >= 0.0f) ? 1.0f : -1.0f;
    U[2]*=d; U[5]*=d; U[8]*=d;        // R = U diag(1,1,d) V^T
    #pragma unroll
    for (int i = 0; i < 3; i++)
        #pragma unroll
        for (int j = 0; j < 3; j++) {
            float s = 0.f;
            #pragma unroll
            for (int k = 0; k < 3; k++) s += U[i*3+k] * V[j*3+k];
            R[i*3+j] = s;
        }
}

// ------------------------------------------------- K1: pose fit per (it,b)

__global__ void fit_poses(const float* __restrict__ src,
                          const float* __restrict__ trg,
                          const float* __restrict__ wts,
                          const float* __restrict__ Tsv,
                          const int*   __restrict__ dimp,
                          float* __restrict__ rows,
                          float* __restrict__ tcam49) {
    int task = blockIdx.x * blockDim.x + threadIdx.x;
    if (task >= NITERS * BB) return;
    int it = task / BB;
    int b  = task % BB;
    int dim = *dimp;

    float Ts[12];
    #pragma unroll
    for (int i = 0; i < 12; i++) Ts[i] = Tsv[i];
    float Vi[12];                      // T_v_s = inv(T_s_v), top 3x4
    #pragma unroll
    for (int i = 0; i < 3; i++)
        #pragma unroll
        for (int j = 0; j < 3; j++) Vi[i*4+j] = Ts[j*4+i];
    #pragma unroll
    for (int i = 0; i < 3; i++)
        Vi[i*4+3] = -(Vi[i*4+0]*Ts[3] + Vi[i*4+1]*Ts[7] + Vi[i*4+2]*Ts[11]);

    float ps[NRANSAC][3], pt[NRANSAC][3], wv[NRANSAC];
    float wsum = 0.f;
    const size_t sb = (size_t)b * 4 * NP;
    for (int k = 0; k < NRANSAC; k++) {
        unsigned h = hash32((unsigned)(task * NRANSAC + k) + 0x9E3779B9u);
        int idx = (int)(h % NP);
        float s0 = src[sb + 0*NP + idx], s1 = src[sb + 1*NP + idx];
        float s2 = src[sb + 2*NP + idx], s3 = src[sb + 3*NP + idx];
        float t0 = trg[sb + 0*NP + idx], t1 = trg[sb + 1*NP + idx];
        float t2 = trg[sb + 2*NP + idx], t3 = trg[sb + 3*NP + idx];
        #pragma unroll
        for (int i = 0; i < 3; i++) {
            ps[k][i] = Vi[i*4]*s0 + Vi[i*4+1]*s1 + Vi[i*4+2]*s2 + Vi[i*4+3]*s3;
            pt[k][i] = Vi[i*4]*t0 + Vi[i*4+1]*t1 + Vi[i*4+2]*t2 + Vi[i*4+3]*t3;
        }
        float wk = wts[(size_t)b * NP + idx];
        wv[k] = wk; wsum += wk;
    }
    float inv = 1.0f / wsum;
    float ms[3] = {0,0,0}, mt[3] = {0,0,0};
    for (int k = 0; k < NRANSAC; k++) {
        float wn = wv[k] * inv;
        #pragma unroll
        for (int i = 0; i < 3; i++) { ms[i] += wn * ps[k][i]; mt[i] += wn * pt[k][i]; }
    }
    float Wm[9] = {0,0,0, 0,0,0, 0,0,0};
    for (int k = 0; k < NRANSAC; k++) {
        float wn = wv[k] * inv;
        #pragma unroll
        for (int i = 0; i < 3; i++) {
            float tci = (pt[k][i] - mt[i]) * wn;
            #pragma unroll
            for (int j = 0; j < 3; j++) Wm[i*3+j] += tci * (ps[k][j] - ms[j]);
        }
    }
    float R[9];
    svd_rigid(Wm, R);
    float Tts[12];
    #pragma unroll
    for (int i = 0; i < 3; i++) {
        #pragma unroll
        for (int j = 0; j < 3; j++) Tts[i*4+j] = R[i*3+j];
        Tts[i*4+3] = mt[i] - (R[i*3]*ms[0] + R[i*3+1]*ms[1] + R[i*3+2]*ms[2]);
    }
    float Tmp[12], Tc[12];
    mul34(Tts, Vi, Tmp);     // T_trg_src @ T_v_s
    mul34(Ts,  Tmp, Tc);     // T_cam = T_s_v @ (...)

    float* rp = rows + (size_t)task * 12;
    if (dim == 2) {
        #pragma unroll
        for (int j = 0; j < 4; j++) {
            rp[j]     = FCAL * Tc[j]     + CUX * Tc[8+j];  // row0 of P @ T_cam
            rp[4+j]   = FCAL * Tc[4+j]   + CVY * Tc[8+j];  // row1 of P @ T_cam
            rp[8+j]   = Tc[8+j];                            // z row of T_cam
        }
    } else {
        #pragma unroll
        for (int j = 0; j < 12; j++) rp[j] = Tc[j];
    }
    if (it == NITERS - 1) {
        float* op = tcam49 + b * 16;
        #pragma unroll
        for (int j = 0; j < 12; j++) op[j] = Tc[j];
        op[12] = 0.f; op[13] = 0.f; op[14] = 0.f; op[15] = 1.f;
    }
}

// ------------------------------- K2: WMMA scoring, 4 poses per instruction

__global__ void score_kernel(const float* __restrict__ src,
                             const float* __restrict__ trg,
                             const float* __restrict__ kp2d,
                             const unsigned char* __restrict__ vsrc,
                             const unsigned char* __restrict__ vtrg,
                             const int*   __restrict__ dimp,
                             const float* __restrict__ rows,
                             int*      __restrict__ counts,
                             unsigned* __restrict__ bad) {
    const int b    = blockIdx.x;
    const int g    = blockIdx.y;          // iteration group: iters 4g..4g+3
    const int lane = threadIdx.x & 31;
    const int wv   = threadIdx.x >> 5;
    const int half = lane >> 4;
    const int m    = lane & 15;
    const int dim  = *dimp;

    // A fragment (16x4 f32): lanes 0-15 hold K{0,1}, lanes 16-31 hold K{2,3}
    // rows 0-2 -> pose j0 {M0,M1,Tz}; 3-5 -> j1; 8-10 -> j2; 11-13 -> j3
    float a0 = 0.f, a1 = 0.f;
    int itl = -1, comp = 0;
    if      (m < 3)            { itl = 0; comp = m;      }
    else if (m < 6)            { itl = 1; comp = m - 3;  }
    else if (m >= 8 && m < 11) { itl = 2; comp = m - 8;  }
    else if (m >= 11 && m < 14){ itl = 3; comp = m - 11; }
    int itRow = 4 * g + itl;
    if (itl >= 0 && itRow < NITERS) {
        const float* rp = rows + ((size_t)itRow * BB + b) * 12 + comp * 4;
        a0 = rp[half * 2 + 0];
        a1 = rp[half * 2 + 1];
    }
    v2f av = {a0, a1};

    const int itA = 4 * g + 2 * half;     // this lane's poses in acc[0..2]/[3..5]
    const int itB = itA + 1;

    int cntA = 0, cntB = 0;
    int nfA = 0, nfB = 0;

    const size_t sb = (size_t)b * 4 * NP;
    const size_t vb = (size_t)b * NP;
    const size_t kb = (size_t)b * 2 * NP;
    // branchless B-fragment row offset: half 0 -> row 0 (x), half 1 -> row 2 (z)
    const size_t rowOff = (size_t)(half * 2) * NP;

    if (dim == 2) {
        #pragma unroll 4
        for (int grp = wv; grp < NP / 16; grp += 8) {
            int n = grp * 16 + m;
            float b0  = src[sb + rowOff + n];       // x (half0) or z (half1)
            float b1y = src[sb + NP + n];           // y row, unconditionally
            float b1  = half ? 1.0f : b1y;          // v_cndmask, no branch
            v2f bv = {b0, b1};
            __builtin_prefetch(&src[sb + rowOff + n + 512], 0, 1);

            v8f acc = {};
            acc = __builtin_amdgcn_wmma_f32_16x16x4_f32(
                false, av, false, bv, (short)0, acc, false, false);

            int val = (int)vsrc[vb + n] & (int)vtrg[vb + n];   // branchless
            float u2 = kp2d[kb + n];
            float v2 = kp2d[kb + NP + n];

            // (d0/z-u)^2 + (d1/z-v)^2 < 16  <=>  (d0-u z)^2 + (d1-v z)^2 < 16 z^2
            float zA  = acc[2];
            float duA = fmaf(-u2, zA, acc[0]);
            float dvA = fmaf(-v2, zA, acc[1]);
            float numA = duA * duA + dvA * dvA;
            float thrA = ERRTOL2 * zA * zA;
            cntA += (int)((numA < thrA) & (val != 0));
            nfA  |= (int)((zA == 0.0f) | !__builtin_isfinite(numA));

            float zB  = acc[5];
            float duB = fmaf(-u2, zB, acc[3]);
            float dvB = fmaf(-v2, zB, acc[4]);
            float numB = duB * duB + dvB * dvB;
            float thrB = ERRTOL2 * zB * zB;
            cntB += (int)((numB < thrB) & (val != 0));
            nfB  |= (int)((zB == 0.0f) | !__builtin_isfinite(numB));
        }
    } else {
        #pragma unroll 4
        for (int grp = wv; grp < NP / 16; grp += 8) {
            int n = grp * 16 + m;
            float b0  = src[sb + rowOff + n];
            float b1y = src[sb + NP + n];
            float b1  = half ? 1.0f : b1y;
            v2f bv = {b0, b1};

            v8f acc = {};
            acc = __builtin_amdgcn_wmma_f32_16x16x4_f32(
                false, av, false, bv, (short)0, acc, false, false);

            int val = (int)vsrc[vb + n] & (int)vtrg[vb + n];
            float tx = trg[sb + 0*NP + n];
            float ty = trg[sb + 1*NP + n];
            float tz = trg[sb + 2*NP + n];

            float dxA = acc[0]-tx, dyA = acc[1]-ty, dzA = acc[2]-tz;
            float e2A = dxA*dxA + dyA*dyA + dzA*dzA;
            cntA += (int)((e2A < ERRTOL2) & (val != 0));
            nfA  |= (int)(!__builtin_isfinite(e2A));

            float dxB = acc[3]-tx, dyB = acc[4]-ty, dzB = acc[5]-tz;
            float e2B = dxB*dxB + dyB*dyB + dzB*dzB;
            cntB += (int)((e2B < ERRTOL2) & (val != 0));
            nfB  |= (int)(!__builtin_isfinite(e2B));
        }
    }

    // reduce within each 16-lane half (xor offsets < 16 stay inside the half)
    for (int off = 8; off >= 1; off >>= 1) {
        cntA += __shfl_xor(cntA, off, 32);
        cntB += __shfl_xor(cntB, off, 32);
        nfA  |= __shfl_xor(nfA,  off, 32);
        nfB  |= __shfl_xor(nfB,  off, 32);
    }
    if (m == 0) {
        if (itA < NITERS) {
            atomicAdd(&counts[itA * BB + b], cntA);
            if (nfA) atomicOr(&bad[itA], 1u);      // reference's bad is global over batches
        }
        if (itB < NITERS) {
            atomicAdd(&counts[itB * BB + b], cntB);
            if (nfB) atomicOr(&bad[itB], 1u);
        }
    }
}

// ---------------------- K3: strict-> argmax scan per batch + emit T_cam(49)

__global__ void select_kernel(const int* __restrict__ counts,
                              const unsigned* __restrict__ bad,
                              const float* __restrict__ tcam49,
                              int* __restrict__ bestArr,
                              float* __restrict__ out) {
    int b = threadIdx.x;
    if (b < BB) {
        int best = -1, maxc = 0;
        for (int i = 0; i < NITERS; i++) {
            int c = counts[i * BB + b];
            if (!bad[i] && c > maxc) { maxc = c; best = i; }
        }
        bestArr[b] = best;
        #pragma unroll
        for (int j = 0; j < 16; j++)
            out[(size_t)BB * NP + b * 16 + j] = tcam49[b * 16 + j];
    }
}

// ------------------------------------ K4: recompute winning mask, emit 0/1

__global__ void emit_kernel(const float* __restrict__ src,
                            const float* __restrict__ trg,
                            const float* __restrict__ kp2d,
                            const unsigned char* __restrict__ vsrc,
                            const unsigned char* __restrict__ vtrg,
                            const int*   __restrict__ dimp,
                            const float* __restrict__ rows,
                            const int*   __restrict__ bestArr,
                            float* __restrict__ out) {
    int b = blockIdx.x;
    int best = bestArr[b];
    int dim = *dimp;
    float r0[4] = {0,0,0,0}, r1[4] = {0,0,0,0}, r2[4] = {0,0,0,0};
    if (best >= 0) {
        const float* rp = rows + ((size_t)best * BB + b) * 12;
        #pragma unroll
        for (int j = 0; j < 4; j++) { r0[j] = rp[j]; r1[j] = rp[4+j]; r2[j] = rp[8+j]; }
    }
    const size_t sb = (size_t)b * 4 * NP;
    for (int n = threadIdx.x; n < NP; n += blockDim.x) {
        float o = 0.f;
        if (best >= 0) {
            float x = src[sb + n], y = src[sb + NP + n];
            float z = src[sb + 2*NP + n], w = src[sb + 3*NP + n];
            float d0 = r0[0]*x + r0[1]*y + r0[2]*z + r0[3]*w;
            float d1 = r1[0]*x + r1[1]*y + r1[2]*z + r1[3]*w;
            float d2 = r2[0]*x + r2[1]*y + r2[2]*z + r2[3]*w;
            int val = (int)vsrc[(size_t)b*NP + n] & (int)vtrg[(size_t)b*NP + n];
            bool inl;
            if (dim == 2) {
                float u2 = kp2d[(size_t)b*2*NP + n], v2 = kp2d[(size_t)b*2*NP + NP + n];
                float du = fmaf(-u2, d2, d0), dv = fmaf(-v2, d2, d1);
                float num = du*du + dv*dv;
                inl = (num < ERRTOL2 * d2 * d2);
            } else {
                float dx = d0 - trg[sb + n], dy = d1 - trg[sb + NP + n];
                float dz = d2 - trg[sb + 2*NP + n];
                inl = (dx*dx + dy*dy + dz*dz < ERRTOL2);
            }
            o = (inl && val) ? 1.0f : 0.0f;
        }
        out[(size_t)b * NP + n] = o;
    }
}

// ---------------------------------------------------------------- launcher

extern "C" void kernel_launch(void* const* d_in, const int* in_sizes, int n_in,
                              void* d_out, int out_size, void* d_ws, size_t ws_size,
                              hipStream_t stream) {
    const float* src  = (const float*)d_in[0];
    const float* trg  = (const float*)d_in[1];
    const float* kp2d = (const float*)d_in[2];
    const unsigned char* vs = (const unsigned char*)d_in[3];
    const unsigned char* vt = (const unsigned char*)d_in[4];
    const float* wts  = (const float*)d_in[5];
    const float* Tsv  = (const float*)d_in[6];
    const int*   dimp = (const int*)d_in[7];
    float* out = (float*)d_out;

    char* ws = (char*)d_ws;
    float*    rows    = (float*)(ws);                 // 50*128*12*4 = 307200 B
    float*    tcam49  = (float*)(ws + 307200);        // 128*16*4    =   8192 B
    int*      counts  = (int*)  (ws + 315392);        // 50*128*4    =  25600 B
    unsigned* bad     = (unsigned*)(ws + 340992);     // 50*4 -> pad =    256 B
    int*      bestArr = (int*)  (ws + 341248);        // 128*4       =    512 B

    hipMemsetAsync(counts, 0, 25600 + 256, stream);   // counts + bad, per call

    fit_poses<<<(NITERS * BB + 63) / 64, 64, 0, stream>>>(
        src, trg, wts, Tsv, dimp, rows, tcam49);

    dim3 grid2(BB, (NITERS + 3) / 4);
    score_kernel<<<grid2, 256, 0, stream>>>(
        src, trg, kp2d, vs, vt, dimp, rows, counts, bad);

    select_kernel<<<1, 128, 0, stream>>>(counts, bad, tcam49, bestArr, out);

    emit_kernel<<<BB, 256, 0, stream>>>(
        src, trg, kp2d, vs, vt, dimp, rows, bestArr, out);
}